// GAT_73967926772327
// MI455X (gfx1250) — compile-verified
//
#include <hip/hip_runtime.h>
#include <hip/hip_bf16.h>
#include <cstdint>
#include <cstddef>

// ---------------- problem constants (match reference) ----------------
#define NNODES  50000
#define NEDGES  800000
#define NE2     850000          // edges + self loops
#define NGRAPHS 128
#define NHEADS  4

typedef float v8f __attribute__((ext_vector_type(8)));
typedef float v2f __attribute__((ext_vector_type(2)));

// ---------------- ordered-uint encoding for float atomicMax ----------------
__device__ __forceinline__ unsigned enc_f32(float f) {
  unsigned u = __float_as_uint(f);
  return (u & 0x80000000u) ? ~u : (u | 0x80000000u);
}
__device__ __forceinline__ float dec_f32(unsigned u) {
  unsigned v = (u & 0x80000000u) ? (u & 0x7FFFFFFFu) : ~u;
  return __uint_as_float(v);
}

// =====================================================================
// GEMM: Hout[M,Nc] = X[M,K] @ W[K,Nc]  using V_WMMA_F32_16X16X4_F32.
// Wave computes a 16x64 tile; block = 8 waves sharing one LDS B-panel.
// B panel stored pair-interleaved: lds[p*PS + 2c + j] = W[kc+2p+j][cbase+c]
// so each WMMA B operand is ONE ds_load_b64 (no register shuffling).
// PS = 160 floats -> pair-row delta mod 64 banks = 32 -> the two half-wave
// K-groups read disjoint bank sets (conflict-free b64 reads).
// =====================================================================
#define KCHUNK 128
#define PS     160   // pair-row stride in floats (64 cols * 2 + 32 pad)

__global__ void __launch_bounds__(256)
gat_gemm_wmma_f32(const float* __restrict__ X, const float* __restrict__ W,
                  float* __restrict__ Hout, int M, int K, int Nc)
{
  extern __shared__ float ldsW[];              // (KCHUNK/2) * PS floats
  const int tid    = threadIdx.x;
  const int lane   = tid & 31;
  const int wid    = tid >> 5;
  const int tile_m = blockIdx.x * 8 + wid;
  const int cbase  = blockIdx.y * 64;
  const bool valid = (tile_m * 16) < M;

  const int m    = lane & 15;
  const int kb   = (lane >> 4) << 1;           // 0 or 2: K-pair base per half-wave
  const int n    = lane & 15;
  const int half = lane >> 4;                  // 0 or 1
  const int row  = tile_m * 16 + m;

  v8f acc0 = {}; v8f acc1 = {}; v8f acc2 = {}; v8f acc3 = {};

  for (int kc = 0; kc < K; kc += KCHUNK) {
    const int KC = min(KCHUNK, K - kc);
    // stage B panel: pairs of rows interleaved per column, float2 DS stores
    for (int i = tid; i < (KC >> 1) * 64; i += 256) {
      const int p = i >> 6;                    // pair index
      const int c = i & 63;                    // column within panel
      const float* wp = &W[(size_t)(kc + 2 * p) * Nc + cbase + c];
      v2f v; v.x = wp[0]; v.y = wp[Nc];
      *reinterpret_cast<v2f*>(&ldsW[p * PS + (c << 1)]) = v;
    }
    __syncthreads();

    if (valid) {
      const float* xrow = X + (size_t)row * K + kc + kb;
      for (int kk = 0; kk < KC; kk += 4) {
        v2f a = *reinterpret_cast<const v2f*>(xrow + kk);     // A[m][kb], A[m][kb+1]
        const float* pb = &ldsW[((kk >> 1) + half) * PS + (n << 1)];
        v2f b0 = *reinterpret_cast<const v2f*>(pb);           // B[kb][n],   B[kb+1][n]
        v2f b1 = *reinterpret_cast<const v2f*>(pb + 32);      // col n+16
        v2f b2 = *reinterpret_cast<const v2f*>(pb + 64);      // col n+32
        v2f b3 = *reinterpret_cast<const v2f*>(pb + 96);      // col n+48
        acc0 = __builtin_amdgcn_wmma_f32_16x16x4_f32(false, a, false, b0, (short)0, acc0, false, false);
        acc1 = __builtin_amdgcn_wmma_f32_16x16x4_f32(false, a, false, b1, (short)0, acc1, false, false);
        acc2 = __builtin_amdgcn_wmma_f32_16x16x4_f32(false, a, false, b2, (short)0, acc2, false, false);
        acc3 = __builtin_amdgcn_wmma_f32_16x16x4_f32(false, a, false, b3, (short)0, acc3, false, false);
      }
    }
    __syncthreads();
  }

  if (valid) {
    // C/D layout: VGPR j -> row j (lanes 0-15) / j+8 (lanes 16-31), col = lane&15
    const int rbase = tile_m * 16 + (half << 3);
    const int c     = cbase + (lane & 15);
    for (int j = 0; j < 8; ++j) {
      float* o = Hout + (size_t)(rbase + j) * Nc + c;
      o[0]  = acc0[j];
      o[16] = acc1[j];
      o[32] = acc2[j];
      o[48] = acc3[j];
    }
  }
}

// ---------------- fills ----------------
__global__ void k_fill_f32(float* p, float v, int n) {
  int i = blockIdx.x * 256 + threadIdx.x;
  if (i < n) p[i] = v;
}
__global__ void k_fill_u32(unsigned* p, unsigned v, int n) {
  int i = blockIdx.x * 256 + threadIdx.x;
  if (i < n) p[i] = v;
}

// ---------------- per-node head scores ----------------
__global__ void k_sdot(const float* __restrict__ h, const float* __restrict__ asrc,
                       const float* __restrict__ adst, float* __restrict__ ssrc,
                       float* __restrict__ sdst, int Fo)
{
  int i = blockIdx.x * 256 + threadIdx.x;          // over N*H
  if (i >= NNODES * NHEADS) return;
  const int nd = i >> 2, hd = i & 3;
  const int HF = NHEADS * Fo;
  const float* hp = h + (size_t)nd * HF + hd * Fo;
  const float* as = asrc + hd * Fo;
  const float* ad = adst + hd * Fo;
  float s0 = 0.f, s1 = 0.f;
  for (int f = 0; f < Fo; ++f) {
    float v = hp[f];
    s0 = fmaf(v, as[f], s0);
    s1 = fmaf(v, ad[f], s1);
  }
  ssrc[i] = s0;
  sdst[i] = s1;
}

__device__ __forceinline__ void edge_sd(const int* __restrict__ ei, int e, int& s, int& d) {
  if (e < NEDGES) { s = ei[e]; d = ei[NEDGES + e]; }
  else            { s = d = e - NEDGES; }            // self loops
}

// ---------------- edge scores + segment max ----------------
__global__ void k_edge_score(const int* __restrict__ ei, const float* __restrict__ ssrc,
                             const float* __restrict__ sdst, float* __restrict__ evalb,
                             unsigned* __restrict__ maxb)
{
  int i = blockIdx.x * 256 + threadIdx.x;           // over NE2*H
  if (i >= NE2 * NHEADS) return;
  const int e = i >> 2, hd = i & 3;
  int s, d; edge_sd(ei, e, s, d);
  float v = ssrc[s * NHEADS + hd] + sdst[d * NHEADS + hd];
  v = (v >= 0.f) ? v : 0.2f * v;                    // leaky relu
  evalb[i] = v;
  atomicMax(&maxb[d * NHEADS + hd], enc_f32(v));
}

// ---------------- exp + segment sum ----------------
__global__ void k_edge_exp(const int* __restrict__ ei, float* __restrict__ evalb,
                           const unsigned* __restrict__ maxb, float* __restrict__ denom)
{
  int i = blockIdx.x * 256 + threadIdx.x;
  if (i >= NE2 * NHEADS) return;
  const int e = i >> 2, hd = i & 3;
  int s, d; edge_sd(ei, e, s, d); (void)s;
  const float mx = dec_f32(maxb[d * NHEADS + hd]);
  const float ex = __expf(evalb[i] - mx);
  evalb[i] = ex;
  atomicAdd(&denom[d * NHEADS + hd], ex);
}

// ---------------- weighted message scatter-add ----------------
__global__ void k_aggregate(const int* __restrict__ ei, const float* __restrict__ h,
                            const float* __restrict__ evalb, const float* __restrict__ denom,
                            float* __restrict__ out, int HF, int hfshift, int foshift)
{
  int i = blockIdx.x * 256 + threadIdx.x;           // over NE2*HF (<2^31)
  if (i >= NE2 * HF) return;
  const int e = i >> hfshift;
  const int f = i & (HF - 1);
  const int hd = f >> foshift;
  int s, d; edge_sd(ei, e, s, d);
  const float alpha = evalb[e * NHEADS + hd] / (denom[d * NHEADS + hd] + 1e-16f);
  atomicAdd(&out[(size_t)d * HF + f], h[(size_t)s * HF + f] * alpha);
}

// ---------------- bias + relu ----------------
__global__ void k_bias_relu(float* __restrict__ out, const float* __restrict__ b, int HF, int total)
{
  int i = blockIdx.x * 256 + threadIdx.x;
  if (i >= total) return;
  const int f = i & (HF - 1);
  out[i] = fmaxf(out[i] + b[f], 0.f);
}

// ---------------- per-graph pooling (sum / count / max) ----------------
__global__ void k_pool_acc(const float* __restrict__ act, const int* __restrict__ batch,
                           float* __restrict__ psum, unsigned* __restrict__ pmax,
                           float* __restrict__ pcnt)
{
  int i = blockIdx.x * 256 + threadIdx.x;           // over N*256
  if (i >= NNODES * 256) return;
  const int nd = i >> 8, f = i & 255;
  const int g = batch[nd];
  const float v = act[(size_t)nd * 256 + f];        // post-relu, v >= 0
  atomicAdd(&psum[g * 256 + f], v);
  atomicMax(&pmax[g * 256 + f], __float_as_uint(v));// valid for nonneg floats
  if (f == 0) atomicAdd(&pcnt[g], 1.0f);
}

// ---------------- readout: [G,512] @ [512,10] + bout ----------------
__global__ void k_readout(const float* __restrict__ psum, const unsigned* __restrict__ pmax,
                          const float* __restrict__ pcnt, const float* __restrict__ Wout,
                          const float* __restrict__ bout, float* __restrict__ out)
{
  int i = blockIdx.x * 256 + threadIdx.x;           // over G*10
  if (i >= NGRAPHS * 10) return;
  const int g = i / 10, j = i % 10;
  const float inv = 1.0f / fmaxf(pcnt[g], 1.0f);
  float acc = bout[j];
  for (int k = 0; k < 256; ++k)
    acc = fmaf(psum[g * 256 + k] * inv, Wout[k * 10 + j], acc);
  for (int k = 0; k < 256; ++k)
    acc = fmaf(__uint_as_float(pmax[g * 256 + k]), Wout[(256 + k) * 10 + j], acc);
  out[i] = acc;
}

// =====================================================================
// host-side orchestration
// =====================================================================
static inline int cdiv(long long a, long long b) { return (int)((a + b - 1) / b); }

extern "C" void kernel_launch(void* const* d_in, const int* in_sizes, int n_in,
                              void* d_out, int out_size, void* d_ws, size_t ws_size,
                              hipStream_t stream)
{
  (void)in_sizes; (void)n_in; (void)out_size; (void)ws_size;
  const float* x     = (const float*)d_in[0];
  const int*   ei    = (const int*)  d_in[1];
  const int*   batch = (const int*)  d_in[2];
  const float* W[3]     = { (const float*)d_in[3],  (const float*)d_in[7],  (const float*)d_in[11] };
  const float* asrc[3]  = { (const float*)d_in[4],  (const float*)d_in[8],  (const float*)d_in[12] };
  const float* adst[3]  = { (const float*)d_in[5],  (const float*)d_in[9],  (const float*)d_in[13] };
  const float* bias[3]  = { (const float*)d_in[6],  (const float*)d_in[10], (const float*)d_in[14] };
  const float* Wout  = (const float*)d_in[15];
  const float* bout  = (const float*)d_in[16];
  float* out = (float*)d_out;

  // -------- workspace carve (all sizes multiples of 256 B) --------
  char* ws = (char*)d_ws;
  float*    buf0  = (float*)ws;            ws += (size_t)NNODES * 256 * 4;   // activations / layer out
  float*    buf1  = (float*)ws;            ws += (size_t)NNODES * 256 * 4;   // h = x@W
  float*    ssrc  = (float*)ws;            ws += (size_t)NNODES * NHEADS * 4;
  float*    sdst  = (float*)ws;            ws += (size_t)NNODES * NHEADS * 4;
  float*    evalb = (float*)ws;            ws += (size_t)NE2   * NHEADS * 4;
  unsigned* maxb  = (unsigned*)ws;         ws += (size_t)NNODES * NHEADS * 4;
  float*    denom = (float*)ws;            ws += (size_t)NNODES * NHEADS * 4;
  float*    psum  = (float*)ws;            ws += (size_t)NGRAPHS * 256 * 4;
  unsigned* pmax  = (unsigned*)ws;         ws += (size_t)NGRAPHS * 256 * 4;
  float*    pcnt  = (float*)ws;            ws += 512;

  const int Fin[3] = { 128, 128, 256 };
  const int Fo [3] = { 32,  64,  64  };
  const size_t lds_bytes = (size_t)(KCHUNK / 2) * PS * sizeof(float);  // 40 KB
  const int gemm_gx = cdiv((NNODES + 15) / 16, 8);   // 391

  const float* cur_in = x;
  for (int L = 0; L < 3; ++L) {
    const int fo = Fo[L], fin = Fin[L];
    const int HF = NHEADS * fo;                      // 128 / 256 / 256
    const int hfshift = (HF == 128) ? 7 : 8;
    const int foshift = (fo == 32) ? 5 : 6;

    // 1) h = in @ W  (WMMA f32)
    dim3 ggrid(gemm_gx, HF / 64);
    gat_gemm_wmma_f32<<<ggrid, 256, lds_bytes, stream>>>(cur_in, W[L], buf1, NNODES, fin, HF);

    // 2) head scores
    k_sdot<<<cdiv((long long)NNODES * NHEADS, 256), 256, 0, stream>>>(buf1, asrc[L], adst[L], ssrc, sdst, fo);

    // 3) init accumulators
    k_fill_u32<<<cdiv(NNODES * NHEADS, 256), 256, 0, stream>>>(maxb, 0u, NNODES * NHEADS);
    k_fill_f32<<<cdiv(NNODES * NHEADS, 256), 256, 0, stream>>>(denom, 0.f, NNODES * NHEADS);
    k_fill_f32<<<cdiv((long long)NNODES * HF, 256), 256, 0, stream>>>(buf0, 0.f, NNODES * HF);

    // 4) segment softmax over dst
    k_edge_score<<<cdiv((long long)NE2 * NHEADS, 256), 256, 0, stream>>>(ei, ssrc, sdst, evalb, maxb);
    k_edge_exp  <<<cdiv((long long)NE2 * NHEADS, 256), 256, 0, stream>>>(ei, evalb, maxb, denom);

    // 5) alpha-weighted scatter-add of h[src]
    k_aggregate<<<cdiv((long long)NE2 * HF, 256), 256, 0, stream>>>(ei, buf1, evalb, denom, buf0, HF, hfshift, foshift);

    // 6) bias + relu
    k_bias_relu<<<cdiv((long long)NNODES * HF, 256), 256, 0, stream>>>(buf0, bias[L], HF, NNODES * HF);

    cur_in = buf0;                                   // next layer input
  }

  // -------- pooling + readout (final activations: buf0 [N,256]) --------
  k_fill_f32<<<cdiv(NGRAPHS * 256, 256), 256, 0, stream>>>(psum, 0.f, NGRAPHS * 256);
  k_fill_u32<<<cdiv(NGRAPHS * 256, 256), 256, 0, stream>>>(pmax, 0u, NGRAPHS * 256);
  k_fill_f32<<<1, 256, 0, stream>>>(pcnt, 0.f, NGRAPHS);
  k_pool_acc<<<cdiv((long long)NNODES * 256, 256), 256, 0, stream>>>(buf0, batch, psum, pmax, pcnt);
  k_readout<<<cdiv(NGRAPHS * 10, 256), 256, 0, stream>>>(psum, pmax, pcnt, Wout, bout, out);
}